// ContrastiveSSL_8435315769539
// MI455X (gfx1250) — compile-verified
//
#include <hip/hip_runtime.h>
#include <hip/hip_bf16.h>

typedef __attribute__((ext_vector_type(2))) float v2f;
typedef __attribute__((ext_vector_type(8))) float v8f;
typedef unsigned long long u64;

#define Nn 8
#define Cc 128
#define Ll 1024
#define Kk 100
#define C_STRIDE 1025           // c is (N, C, L+1)
#define C_PER_N (Cc * C_STRIDE) // 131200
#define Z_PER_N (Cc * Ll)       // 131072
#define FNV_OFF 1469598103934665603ULL
#define FNV_PRM 1099511628211ULL

__device__ __forceinline__ unsigned canon_bits(float v) {
    unsigned b = __float_as_uint(v);
    if ((b & 0x7FFFFFFFu) == 0u) b = 0u;   // -0.0 == +0.0 under float ==
    return b;
}

// ---------------- Phase 1: per-row norms + equality hashes ----------------
__global__ void ssl_prep(const float* __restrict__ cc, const float* __restrict__ zz,
                         float* __restrict__ cnorm, float* __restrict__ znorm,
                         u64* __restrict__ hc, u64* __restrict__ hz) {
    int id = blockIdx.x * blockDim.x + threadIdx.x;   // 0..8191 = n*1024+t
    if (id >= Nn * Ll) return;
    int n = id >> 10, t = id & 1023;
    {   // z_seq row t of batch n: z[n][ch][t], stride Ll — coalesced across t
        const float* p = zz + (size_t)n * Z_PER_N + t;
        float ss = 0.f; u64 h = FNV_OFF;
        #pragma unroll 8
        for (int ch = 0; ch < Cc; ++ch) {
            float v = p[ch * Ll];
            ss = fmaf(v, v, ss);
            h = (h ^ (u64)canon_bits(v)) * FNV_PRM;
        }
        znorm[id] = fmaxf(sqrtf(ss), 1e-8f);
        hz[id] = h;
    }
    {   // c_seq row t: c[n][ch][t+1], stride 1025
        const float* p = cc + (size_t)n * C_PER_N + t + 1;
        float ss = 0.f; u64 h = FNV_OFF;
        #pragma unroll 8
        for (int ch = 0; ch < Cc; ++ch) {
            float v = p[ch * C_STRIDE];
            ss = fmaf(v, v, ss);
            h = (h ^ (u64)canon_bits(v)) * FNV_PRM;
        }
        cnorm[id] = fmaxf(sqrtf(ss), 1e-8f);
        hc[id] = h;
    }
}

// ---------------- Phase 2: G[n] = C_seq[n] @ Z_seq[n]^T via f32 WMMA -------
// One wave computes a 32x32 tile of G. K=128 in steps of 4 -> 128 WMMAs/tile.
// A-frag (16x4 f32, 2 VGPRs): lanes 0-15 hold K=0(v0)/K=1(v1), lanes 16-31 K=2/K=3.
// B-frag (4x16): symmetric. C/D: VGPR r -> M = r + 8*(lane>=16), N = lane&15.
__global__ void ssl_gemm(const float* __restrict__ cc, const float* __restrict__ zz,
                         float* __restrict__ G) {
    int lane = threadIdx.x & 31;
    int tileId = blockIdx.x * (blockDim.x >> 5) + (threadIdx.x >> 5);
    if (tileId >= Nn * 32 * 32) return;
    int n = tileId >> 10;
    int rem = tileId & 1023;
    int trow0 = (rem >> 5) << 5;      // t-tile origin
    int ucol0 = (rem & 31) << 5;      // u-tile origin
    int lo = lane & 15;
    int hi = lane >> 4;               // 0 or 1

    const float* Cbase = cc + (size_t)n * C_PER_N + 1 + trow0 + lo;  // (ch,t) -> Cbase[ch*1025 + dt]
    const float* Zbase = zz + (size_t)n * Z_PER_N + ucol0 + lo;      // (ch,u) -> Zbase[ch*1024 + du]

    v8f acc00 = {}, acc01 = {}, acc10 = {}, acc11 = {};

    #pragma unroll 4
    for (int s = 0; s < 32; ++s) {
        int chA = 4 * s + 2 * hi;
        v2f a0, a1, b0, b1;
        a0.x = Cbase[chA * C_STRIDE];           a0.y = Cbase[(chA + 1) * C_STRIDE];
        a1.x = Cbase[chA * C_STRIDE + 16];      a1.y = Cbase[(chA + 1) * C_STRIDE + 16];
        b0.x = Zbase[chA * Ll];                 b0.y = Zbase[(chA + 1) * Ll];
        b1.x = Zbase[chA * Ll + 16];            b1.y = Zbase[(chA + 1) * Ll + 16];
        acc00 = __builtin_amdgcn_wmma_f32_16x16x4_f32(false, a0, false, b0, (short)0, acc00, false, false);
        acc01 = __builtin_amdgcn_wmma_f32_16x16x4_f32(false, a0, false, b1, (short)0, acc01, false, false);
        acc10 = __builtin_amdgcn_wmma_f32_16x16x4_f32(false, a1, false, b0, (short)0, acc10, false, false);
        acc11 = __builtin_amdgcn_wmma_f32_16x16x4_f32(false, a1, false, b1, (short)0, acc11, false, false);
    }

    float* Gn = G + ((size_t)n << 20);
    #pragma unroll
    for (int r = 0; r < 8; ++r) {
        int t0 = trow0 + r + 8 * hi;
        Gn[(size_t)t0 * Ll + ucol0 + lo]             = acc00[r];
        Gn[(size_t)t0 * Ll + ucol0 + 16 + lo]        = acc01[r];
        Gn[(size_t)(t0 + 16) * Ll + ucol0 + lo]      = acc10[r];
        Gn[(size_t)(t0 + 16) * Ll + ucol0 + 16 + lo] = acc11[r];
    }
}

// ---------------- Phase 3: gather + normalize + mask ----------------------
__global__ void ssl_logits(const float* __restrict__ cc, const float* __restrict__ zz,
                           const int* __restrict__ neg, const float* __restrict__ G,
                           const float* __restrict__ cnorm, const float* __restrict__ znorm,
                           const u64* __restrict__ hc, const u64* __restrict__ hz,
                           float* __restrict__ out) {
    int wid = (blockIdx.x * blockDim.x + threadIdx.x) >> 5;  // one wave per (n,t)
    int lane = threadIdx.x & 31;
    if (wid >= Nn * Ll) return;
    int n = wid >> 10, t = wid & 1023;
    float cn = cnorm[wid];
    u64 h = hc[wid];
    const float* grow   = G + ((size_t)wid << 10);
    const int*   negrow = neg + (size_t)wid * Kk;
    float*       orow   = out + (size_t)wid * (Kk + 1);
    #pragma unroll
    for (int kk = 0; kk < 4; ++kk) {
        int k = kk * 32 + lane;
        if (k > Kk) continue;
        int u = (k == 0) ? t : negrow[k - 1];
        float g  = grow[u];
        float tn = znorm[(n << 10) + u];
        float logit = g / (cn * tn) * 2.0f;   // /TEMP, TEMP=0.5 exact
        if (k > 0 && hz[(n << 10) + u] == h) {
            // rare: verify exact element-wise equality of c_seq[n,t,:] vs z_seq[n,u,:]
            bool eq = true;
            const float* cp = cc + (size_t)n * C_PER_N + t + 1;
            const float* zp = zz + (size_t)n * Z_PER_N + u;
            for (int ch = 0; ch < Cc; ++ch)
                eq = eq && (cp[ch * C_STRIDE] == zp[ch * Ll]);
            if (eq) logit = -__builtin_huge_valf();
        }
        orow[k] = logit;
    }
}

// ---------------- Fallback (ws too small): direct fused path --------------
__global__ void ssl_direct(const float* __restrict__ cc, const float* __restrict__ zz,
                           const int* __restrict__ neg, float* __restrict__ out) {
    int wid = (blockIdx.x * blockDim.x + threadIdx.x) >> 5;
    int lane = threadIdx.x & 31;
    if (wid >= Nn * Ll) return;
    int n = wid >> 10, t = wid & 1023;
    const float* cp = cc + (size_t)n * C_PER_N + t + 1;
    float c0 = cp[(lane * 4 + 0) * C_STRIDE], c1 = cp[(lane * 4 + 1) * C_STRIDE];
    float c2 = cp[(lane * 4 + 2) * C_STRIDE], c3 = cp[(lane * 4 + 3) * C_STRIDE];
    float css = c0 * c0 + c1 * c1 + c2 * c2 + c3 * c3;
    for (int off = 16; off; off >>= 1) css += __shfl_xor(css, off, 32);
    float cn = fmaxf(sqrtf(css), 1e-8f);
    const int* negrow = neg + (size_t)wid * Kk;
    float* orow = out + (size_t)wid * (Kk + 1);
    const float* zb = zz + (size_t)n * Z_PER_N;
    for (int k = 0; k <= Kk; ++k) {
        int u = k ? negrow[k - 1] : t;
        float z0 = zb[(lane * 4 + 0) * Ll + u], z1 = zb[(lane * 4 + 1) * Ll + u];
        float z2 = zb[(lane * 4 + 2) * Ll + u], z3 = zb[(lane * 4 + 3) * Ll + u];
        float dot = c0 * z0 + c1 * z1 + c2 * z2 + c3 * z3;
        float zss = z0 * z0 + z1 * z1 + z2 * z2 + z3 * z3;
        int eq = (c0 == z0) & (c1 == z1) & (c2 == z2) & (c3 == z3);
        for (int off = 16; off; off >>= 1) {
            dot += __shfl_xor(dot, off, 32);
            zss += __shfl_xor(zss, off, 32);
            eq  &= __shfl_xor(eq, off, 32);
        }
        if (lane == 0) {
            float tn = fmaxf(sqrtf(zss), 1e-8f);
            float l = dot / (cn * tn) * 2.0f;
            if (k && eq) l = -__builtin_huge_valf();
            orow[k] = l;
        }
    }
}

extern "C" void kernel_launch(void* const* d_in, const int* in_sizes, int n_in,
                              void* d_out, int out_size, void* d_ws, size_t ws_size,
                              hipStream_t stream) {
    const float* z   = (const float*)d_in[0];   // (8,128,1024)
    const float* c   = (const float*)d_in[1];   // (8,128,1025)
    const int*   neg = (const int*)d_in[2];     // (8,1024,100)
    float* out = (float*)d_out;                 // (8192,101)

    const size_t gElems = (size_t)Nn * Ll * Ll;                       // 8M floats
    const size_t needed = gElems * 4 + 2 * 8192 * 4 + 2 * 8192 * 8;   // ~33.8 MB

    if (ws_size >= needed) {
        float* G  = (float*)d_ws;
        float* cn = G + gElems;
        float* zn = cn + Nn * Ll;
        u64*   hc = (u64*)(zn + Nn * Ll);
        u64*   hz = hc + Nn * Ll;
        ssl_prep  <<<(Nn * Ll + 255) / 256, 256, 0, stream>>>(c, z, cn, zn, hc, hz);
        ssl_gemm  <<<(Nn * 32 * 32) / 4, 128, 0, stream>>>(c, z, G);   // 4 waves/block
        ssl_logits<<<(Nn * Ll) / 8, 256, 0, stream>>>(c, z, neg, G, cn, zn, hc, hz, out);
    } else {
        ssl_direct<<<(Nn * Ll) / 8, 256, 0, stream>>>(c, z, neg, out);
    }
}